// CountingLoss_23828478558431
// MI455X (gfx1250) — compile-verified
//
#include <hip/hip_runtime.h>
#include <cstdint>
#include <cstddef>

// Problem constants (match reference: B=16, C=2, H=W=1024, 32 CCL iters)
#define IMG_W   1024
#define IMG_H   1024
#define HW      (IMG_W * IMG_H)          // 1,048,576 pixels per image
#define NBATCH  16
#define PRES_WORDS (HW / 32)             // 32768 u32 words of presence bits per image

#define TILE    64
#define HALO    32                       // 32 iterations propagate <= 32 px
#define REG     (TILE + 2 * HALO)        // 128
#define RE2     (REG * REG)              // 16384 cells -> 64 KB u32, x2 ping-pong = 128 KB LDS
#define CCL_ITERS 32

#define THREADS 256
#define SEG_BLOCKS 4096

// ---------------- WMMA wave32 sum reduction (V_WMMA_F32_16X16X4_F32) -------
typedef float v2f __attribute__((ext_vector_type(2)));
typedef float v8f __attribute__((ext_vector_type(8)));

// Sum one float per lane across a full wave32. Requires EXEC == all ones.
// A is 16x4 f32 (2 VGPRs/lane), B = ones (4x16), D[m][n] = rowsum(A,m).
__device__ __forceinline__ float wave_sum32(float v) {
  v2f a; a.x = v;    a.y = 0.0f;
  v2f b; b.x = 1.0f; b.y = 1.0f;
  v8f c = {};
  c = __builtin_amdgcn_wmma_f32_16x16x4_f32(false, a, false, b, (short)0, c,
                                            false, false);
  float s = c[0] + c[1] + c[2] + c[3] + c[4] + c[5] + c[6] + c[7];
  return __shfl(s, 0, 32) + __shfl(s, 16, 32);
}

// Block-wide (256 threads = 8 waves) sum; valid in threadIdx.x == 0.
__device__ __forceinline__ float block_sum256(float v) {
  __shared__ float smem[8];
  float w = wave_sum32(v);
  const int lane = threadIdx.x & 31;
  const int wid  = threadIdx.x >> 5;
  if (lane == 0) smem[wid] = w;
  __syncthreads();
  float r = 0.0f;
  if (wid == 0) {                        // whole wave 0 enters: EXEC all ones
    float x = (lane < 8) ? smem[lane] : 0.0f;
    r = wave_sum32(x);
  }
  return r;
}

__device__ __forceinline__ unsigned int umax2(unsigned int a, unsigned int b) {
  return a > b ? a : b;
}

// Generic (__shared__) pointer -> 32-bit LDS byte offset for async-DMA asm.
__device__ __forceinline__ unsigned int lds_offset_of(const void* p) {
  return (unsigned int)(uintptr_t)(__attribute__((address_space(3))) const void*)p;
}

// CDNA5 async global->LDS copy (GV mode, per-lane, EXEC-masked, ASYNCcnt).
__device__ __forceinline__ void async_copy_b32_to_lds(unsigned int lds_addr,
                                                      const void* gptr) {
  asm volatile("global_load_async_to_lds_b32 %0, %1, off"
               :: "v"(lds_addr), "v"((unsigned long long)(uintptr_t)gptr)
               : "memory");
}

__device__ __forceinline__ void wait_asynccnt0() {
  asm volatile("s_wait_asynccnt 0x0" ::: "memory");
}

// ---------------- workspace zeroing (presence bitmaps) ---------------------
__global__ void zero_kernel(unsigned int* __restrict__ p, size_t n) {
  size_t i = (size_t)blockIdx.x * blockDim.x + threadIdx.x;
  const size_t stride = (size_t)gridDim.x * blockDim.x;
  for (; i < n; i += stride) p[i] = 0u;
}

// ---------------- seg loss: per-pixel 2-class log-softmax CE ---------------
__device__ __forceinline__ float ce_elem(float p0, float p1, float t) {
  const float tb  = (t > 0.5f) ? 1.0f : 0.0f;
  const float m   = fmaxf(p0, p1);
  const float lse = m + __logf(__expf(p0 - m) + __expf(p1 - m));
  return tb * (p1 - lse) + (1.0f - tb) * (p0 - lse);
}

__global__ __launch_bounds__(THREADS)
void seg_loss_kernel(const float* __restrict__ pred,
                     const float* __restrict__ target,
                     float* __restrict__ partials) {
  const size_t HV = HW / 4;                        // float4 vectors per image
  const size_t NV = (size_t)NBATCH * HV;
  const size_t stride = (size_t)gridDim.x * blockDim.x;
  float acc = 0.0f;
  for (size_t i = (size_t)blockIdx.x * blockDim.x + threadIdx.x; i < NV;
       i += stride) {
    const size_t b  = i / HV;
    const size_t pv = i - b * HV;
    const float4* c0 = (const float4*)(pred + b * (size_t)(2 * HW));
    const float4* c1 = (const float4*)(pred + b * (size_t)(2 * HW) + HW);
    const float4 p0 = c0[pv];
    const float4 p1 = c1[pv];
    const float4 t4 = ((const float4*)target)[i];
    acc += ce_elem(p0.x, p1.x, t4.x) + ce_elem(p0.y, p1.y, t4.y) +
           ce_elem(p0.z, p1.z, t4.z) + ce_elem(p0.w, p1.w, t4.w);
  }
  const float bs = block_sum256(-acc);
  if (threadIdx.x == 0) partials[blockIdx.x] = bs;
}

// ---------------- fused CCL: binarize + 32 max-pool iters in LDS + scatter -
// One workgroup = one 64x64 output tile with a 32-px halo (128x128 region).
// Shifted labels: fg pixel at flat index f gets label f+1; background = 0.
// Raw f32 staged into Bsh via async-DMA (no VGPR round-trip), then converted.
// Separable 3x3 max: horizontal pass A->B, vertical pass (+fg select) B->A.
__global__ __launch_bounds__(THREADS)
void ccl_kernel(const float* __restrict__ src, long long batch_stride,
                unsigned int* __restrict__ presence) {
  __shared__ unsigned int A[RE2];
  __shared__ unsigned int Bsh[RE2];

  const int b   = blockIdx.z;
  const int ty0 = blockIdx.y * TILE - HALO;
  const int tx0 = blockIdx.x * TILE - HALO;
  const float* img = src + (long long)b * batch_stride;

  // Stage raw pixels into Bsh with async global->LDS copies (in-image lanes only).
  const unsigned int bsh_base = lds_offset_of(Bsh);
  for (int k = threadIdx.x; k < RE2; k += THREADS) {
    const int r = k >> 7;
    const int c = k & (REG - 1);
    const int gy = ty0 + r;
    const int gx = tx0 + c;
    if ((unsigned)gy < (unsigned)IMG_H && (unsigned)gx < (unsigned)IMG_W) {
      async_copy_b32_to_lds(bsh_base + (unsigned)k * 4u,
                            img + (size_t)gy * IMG_W + gx);
    }
  }
  wait_asynccnt0();
  __syncthreads();

  // Binarize + init shifted labels from staged data (OOB halo cells -> 0).
  for (int k = threadIdx.x; k < RE2; k += THREADS) {
    const int r = k >> 7;
    const int c = k & (REG - 1);
    const int gy = ty0 + r;
    const int gx = tx0 + c;
    unsigned int lab = 0u;
    if ((unsigned)gy < (unsigned)IMG_H && (unsigned)gx < (unsigned)IMG_W) {
      if (__uint_as_float(Bsh[k]) > 0.5f)
        lab = (unsigned)(gy * IMG_W + gx) + 1u;
    }
    A[k] = lab;
  }
  __syncthreads();

  for (int it = 0; it < CCL_ITERS; ++it) {
    // horizontal 3-max: A -> Bsh
    for (int k = threadIdx.x; k < RE2; k += THREADS) {
      const int c = k & (REG - 1);
      unsigned int m = A[k];
      const unsigned int l = (c > 0)       ? A[k - 1] : 0u;
      const unsigned int r = (c < REG - 1) ? A[k + 1] : 0u;
      Bsh[k] = umax2(m, umax2(l, r));
    }
    __syncthreads();
    // vertical 3-max + fg select: Bsh -> A
    for (int k = threadIdx.x; k < RE2; k += THREADS) {
      const int r = k >> 7;
      const unsigned int cur = A[k];
      unsigned int m = Bsh[k];
      const unsigned int u = (r > 0)       ? Bsh[k - REG] : 0u;
      const unsigned int d = (r < REG - 1) ? Bsh[k + REG] : 0u;
      m = umax2(m, umax2(u, d));
      A[k] = (cur != 0u) ? m : 0u;       // non-fg stays 0 (label invariant)
    }
    __syncthreads();
  }

  // Scatter presence bits for the exact interior (label>0 after un-shift).
  unsigned int* pres = presence + (size_t)b * PRES_WORDS;
  for (int k = threadIdx.x; k < RE2; k += THREADS) {
    const int r = k >> 7;
    const int c = k & (REG - 1);
    if ((unsigned)(r - HALO) < (unsigned)TILE &&
        (unsigned)(c - HALO) < (unsigned)TILE) {
      const unsigned int v = A[k];
      if (v >= 2u) {                     // shifted label 1 == original label 0 (excluded)
        const unsigned int L = v - 1u;   // original label in [1, HW)
        atomicOr(&pres[L >> 5], 1u << (L & 31u));
      }
    }
  }
}

// ---------------- popcount the presence bitmaps ----------------------------
// blockIdx.x in [0,32): 0..15 = target images, 16..31 = pred images.
__global__ __launch_bounds__(THREADS)
void count_kernel(const unsigned int* __restrict__ presence,
                  float* __restrict__ counts) {
  const uint4* p = (const uint4*)(presence + (size_t)blockIdx.x * PRES_WORDS);
  unsigned int s = 0;
  for (int w = threadIdx.x; w < PRES_WORDS / 4; w += THREADS) {
    const uint4 q = p[w];
    s += __popc(q.x) + __popc(q.y) + __popc(q.z) + __popc(q.w);
  }
  const float bs = block_sum256((float)s);
  if (threadIdx.x == 0) counts[blockIdx.x] = bs;
}

// ---------------- finalize both scalars ------------------------------------
__global__ __launch_bounds__(THREADS)
void finalize_kernel(const float* __restrict__ partials,
                     const float* __restrict__ counts,
                     float* __restrict__ out) {
  float acc = 0.0f;
  for (int i = threadIdx.x; i < SEG_BLOCKS; i += THREADS) acc += partials[i];
  const float total = block_sum256(acc);
  if (threadIdx.x == 0) {
    out[0] = total / (float)((size_t)NBATCH * HW);     // seg_loss (mean)
    float cl = 0.0f;
    for (int b = 0; b < NBATCH; ++b)
      cl += fabsf(counts[NBATCH + b] - counts[b]);     // |pred - target|
    out[1] = cl / (float)NBATCH;                        // count_loss (mean)
  }
}

// ---------------- launch ---------------------------------------------------
extern "C" void kernel_launch(void* const* d_in, const int* in_sizes, int n_in,
                              void* d_out, int out_size, void* d_ws,
                              size_t ws_size, hipStream_t stream) {
  const float* pred   = (const float*)d_in[0];   // (16, 2, 1024, 1024) f32
  const float* target = (const float*)d_in[1];   // (16, 1024, 1024) f32
  float* out = (float*)d_out;                    // [seg_loss, count_loss]

  // Workspace layout: presence bitmaps (4 MB) | counts (32 f32) | partials
  unsigned int* presence = (unsigned int*)d_ws;
  const size_t pres_total = (size_t)2 * NBATCH * PRES_WORDS;  // u32 words
  float* counts   = (float*)((char*)d_ws + pres_total * sizeof(unsigned int));
  float* partials = counts + 32;

  zero_kernel<<<512, THREADS, 0, stream>>>(presence, pres_total);

  seg_loss_kernel<<<SEG_BLOCKS, THREADS, 0, stream>>>(pred, target, partials);

  const dim3 grid(IMG_W / TILE, IMG_H / TILE, NBATCH);  // (16,16,16)
  // target images: batch stride HW
  ccl_kernel<<<grid, THREADS, 0, stream>>>(target, (long long)HW, presence);
  // pred channel 1: base offset +HW, batch stride 2*HW
  ccl_kernel<<<grid, THREADS, 0, stream>>>(pred + HW, (long long)(2 * HW),
                                           presence + (size_t)NBATCH * PRES_WORDS);

  count_kernel<<<2 * NBATCH, THREADS, 0, stream>>>(presence, counts);
  finalize_kernel<<<1, THREADS, 0, stream>>>(partials, counts, out);
}